// PostProcess_89970974916701
// MI455X (gfx1250) — compile-verified
//
#include <hip/hip_runtime.h>
#include <hip/hip_bf16.h>
#include <stdint.h>

#define BATCH 256
#define NQ    900
#define NC    91
#define QCNT  (NQ * NC)          // 81900 logits per batch row
#define BS    512                // 16 wave32 waves per workgroup
#define NBINS 2048               // 11-bit radix histogram
#define CAND_MAX 2048
#define TOPK_PAD 1024

typedef unsigned long long u64;

#define AS1 __attribute__((address_space(1)))
#define AS3 __attribute__((address_space(3)))

typedef int v4i __attribute__((ext_vector_type(4)));

// ---- gfx1250 async global->LDS path (probe-guarded) ---------------------
#if defined(__has_builtin)
#  if __has_builtin(__builtin_amdgcn_global_load_async_to_lds_b128)
#    define HAVE_ASYNC_LDS 1
#  endif
#  if __has_builtin(__builtin_amdgcn_s_wait_asynccnt)
#    define WAIT_ASYNC(n) __builtin_amdgcn_s_wait_asynccnt(n)
#  endif
#endif
#ifndef HAVE_ASYNC_LDS
#  define HAVE_ASYNC_LDS 0
#endif
#ifndef WAIT_ASYNC
#  define WAIT_ASYNC(n) asm volatile("s_wait_asynccnt %0" :: "i"(n) : "memory")
#endif

#if HAVE_ASYNC_LDS
__device__ __forceinline__ void async_load_b128(const float* gsrc, float* lds_dst) {
    // Builtin signature (from hipcc diagnostic): arg0 = AS1 (global) int4*,
    // non-const; arg1 = LDS-side int4*; then imm offset, imm cpol.
    __builtin_amdgcn_global_load_async_to_lds_b128(
        (AS1 v4i*)(const_cast<float*>(gsrc)),
        (AS3 v4i*)(lds_dst), 0, 0);
}
#endif

// Monotonic float -> uint order key (works on raw logits; sigmoid is monotonic)
__device__ __forceinline__ unsigned orderkey(float f) {
    unsigned u = __float_as_uint(f);
    return (u & 0x80000000u) ? ~u : (u | 0x80000000u);
}
__device__ __forceinline__ float key_to_float(unsigned key) {
    unsigned u = (key & 0x80000000u) ? (key & 0x7FFFFFFFu) : ~key;
    return __uint_as_float(u);
}
// 64-bit sort record: key in high 32, inverted index low 32 so that a
// descending sort yields descending score with ascending-index tie-break
// (matches jax.lax.top_k).
__device__ __forceinline__ u64 pack_kv(unsigned key, unsigned idx) {
    return ((u64)key << 32) | (u64)(0xFFFFFFFFu - idx);
}

// In-LDS bitonic sort, descending, n = power of two, all BS threads participate.
__device__ void bitonic_sort_desc(u64* a, int n, int tid) {
    for (int k = 2; k <= n; k <<= 1) {
        for (int j = k >> 1; j > 0; j >>= 1) {
            for (int i = tid; i < n; i += BS) {
                int ixj = i ^ j;
                if (ixj > i) {
                    u64 x = a[i], y = a[ixj];
                    bool desc_seg = ((i & k) == 0);
                    if (desc_seg ? (x < y) : (x > y)) { a[i] = y; a[ixj] = x; }
                }
            }
            __syncthreads();
        }
    }
}

__global__ __launch_bounds__(BS)
void detr_postprocess_topk(const float* __restrict__ logits,
                           const float* __restrict__ boxes_in,
                           const float* __restrict__ tsizes,
                           float* __restrict__ out) {
    const int b   = blockIdx.x;
    const int tid = threadIdx.x;
    const float* __restrict__ src = logits + (size_t)b * QCNT;

    __shared__ unsigned hist[NBINS];        // 8 KB
    __shared__ u64 cand[CAND_MAX];          // 16 KB
    __shared__ u64 topk[TOPK_PAD];          // 8 KB
#if HAVE_ASYNC_LDS
    __shared__ float stage[2][BS * 4];      // 16 KB double buffer
#endif
    __shared__ int s_tb, s_above, s_need, s_candCnt, s_outCnt;

    for (int i = tid; i < NBINS; i += BS) hist[i] = 0;
    if (tid == 0) { s_candCnt = 0; s_outCnt = 0; }
    __syncthreads();

    // ---------------- Phase 1: 11-bit histogram of order keys ----------------
#if HAVE_ASYNC_LDS
    const int CH  = BS * 4;                     // 2048 floats per chunk
    const int nch = (QCNT + CH - 1) / CH;       // 40 chunks
    {   // issue chunk 0
        int nf4 = (QCNT - 0) / 4; if (nf4 > BS) nf4 = BS;
        if (tid < nf4)
            async_load_b128(src + tid * 4, &stage[0][tid * 4]);
    }
    for (int ch = 0; ch < nch; ++ch) {
        const int buf  = ch & 1;
        const int base = ch * CH;
        if (ch + 1 < nch) {                     // issue next chunk, then wait for this one
            const int nbase = (ch + 1) * CH;
            int nf4 = (QCNT - nbase) / 4; if (nf4 > BS) nf4 = BS;
            if (tid < nf4)
                async_load_b128(src + nbase + tid * 4, &stage[buf ^ 1][tid * 4]);
            WAIT_ASYNC(1);
        } else {
            WAIT_ASYNC(0);
        }
        __syncthreads();
        int vc = QCNT - base; if (vc > CH) vc = CH;
        for (int p = tid; p < vc; p += BS)      // stride-BS: conflict-free LDS banks
            atomicAdd(&hist[orderkey(stage[buf][p]) >> 21], 1u);
        __syncthreads();                        // buffer reusable for chunk ch+2
    }
#else
    for (int i4 = tid; i4 < QCNT / 4; i4 += BS) {
        const float4 v = reinterpret_cast<const float4*>(src)[i4];
        atomicAdd(&hist[orderkey(v.x) >> 21], 1u);
        atomicAdd(&hist[orderkey(v.y) >> 21], 1u);
        atomicAdd(&hist[orderkey(v.z) >> 21], 1u);
        atomicAdd(&hist[orderkey(v.w) >> 21], 1u);
    }
    __syncthreads();
#endif

    // ---------------- Phase 2: find threshold bin (suffix scan) --------------
    if (tid == 0) {
        unsigned cum = 0; int tb = 0; unsigned above = 0;
        for (int bin = NBINS - 1; bin >= 0; --bin) {
            unsigned c = hist[bin];
            if (cum + c >= (unsigned)NQ) { tb = bin; above = cum; break; }
            cum += c;
        }
        s_tb = tb; s_above = (int)above; s_need = NQ - (int)above;
    }
    __syncthreads();
    const int tb = s_tb;

    // ------- Phase 3: collect winners (> bin) and tie-bin candidates ---------
    for (int i4 = tid; i4 < QCNT / 4; i4 += BS) {
        if (i4 + BS < QCNT / 4)
            __builtin_prefetch(src + (size_t)(i4 + BS) * 4, 0, 1);  // global_prefetch_b8
        const float4 v = reinterpret_cast<const float4*>(src)[i4];
        const float vv[4] = {v.x, v.y, v.z, v.w};
#pragma unroll
        for (int e = 0; e < 4; ++e) {
            unsigned key = orderkey(vv[e]);
            int bin = (int)(key >> 21);
            unsigned idx = (unsigned)(i4 * 4 + e);
            if (bin > tb) {
                int p = atomicAdd(&s_outCnt, 1);
                topk[p] = pack_kv(key, idx);            // p < 900 guaranteed
            } else if (bin == tb) {
                int p = atomicAdd(&s_candCnt, 1);
                if (p < CAND_MAX) cand[p] = pack_kv(key, idx);
            }
        }
    }
    __syncthreads();

    // ------- Phase 4: sort tie-bin candidates, append needed remainder -------
    int cc = s_candCnt; if (cc > CAND_MAX) cc = CAND_MAX;
    int n = 1; while (n < cc) n <<= 1;
    for (int i = cc + tid; i < n; i += BS) cand[i] = 0;
    __syncthreads();
    bitonic_sort_desc(cand, n, tid);

    for (int r = tid; r < s_need; r += BS) topk[s_above + r] = cand[r];
    for (int r = NQ + tid; r < TOPK_PAD; r += BS) topk[r] = 0;
    __syncthreads();

    // ---------------- Phase 5: final sort of the 900 winners ------------------
    bitonic_sort_desc(topk, TOPK_PAD, tid);

    // ---------------- Phase 6: scores / labels / boxes ------------------------
    float* __restrict__ out_scores = out;
    float* __restrict__ out_labels = out + (size_t)BATCH * NQ;
    float* __restrict__ out_boxes  = out + (size_t)2 * BATCH * NQ;
    const float img_h = tsizes[b * 2 + 0];
    const float img_w = tsizes[b * 2 + 1];

    for (int r = tid; r < NQ; r += BS) {
        u64 e = topk[r];
        unsigned key = (unsigned)(e >> 32);
        unsigned idx = 0xFFFFFFFFu - (unsigned)(e & 0xFFFFFFFFu);
        float lg    = key_to_float(key);
        float score = 1.0f / (1.0f + __expf(-lg));
        int q   = (int)(idx / NC);
        int lab = (int)(idx - (unsigned)q * NC);

        const float4 bx = reinterpret_cast<const float4*>(boxes_in)[(size_t)b * NQ + q];
        float x0 = (bx.x - 0.5f * bx.z) * img_w;
        float y0 = (bx.y - 0.5f * bx.w) * img_h;
        float x1 = (bx.x + 0.5f * bx.z) * img_w;
        float y1 = (bx.y + 0.5f * bx.w) * img_h;

        out_scores[(size_t)b * NQ + r] = score;
        out_labels[(size_t)b * NQ + r] = (float)lab;
        float* ob = out_boxes + ((size_t)b * NQ + r) * 4;
        ob[0] = x0; ob[1] = y0; ob[2] = x1; ob[3] = y1;
    }
}

extern "C" void kernel_launch(void* const* d_in, const int* in_sizes, int n_in,
                              void* d_out, int out_size, void* d_ws, size_t ws_size,
                              hipStream_t stream) {
    (void)in_sizes; (void)n_in; (void)d_ws; (void)ws_size; (void)out_size;
    const float* pred_logits  = (const float*)d_in[0];
    const float* pred_boxes   = (const float*)d_in[1];
    const float* target_sizes = (const float*)d_in[2];
    float* out = (float*)d_out;
    detr_postprocess_topk<<<dim3(BATCH), dim3(BS), 0, stream>>>(
        pred_logits, pred_boxes, target_sizes, out);
}